// simLayer_29411936043189
// MI455X (gfx1250) — compile-verified
//
#include <hip/hip_runtime.h>

// ---------------- problem constants ----------------
#define NB 4
#define NH 128      // H+2
#define NW 128      // W+2
#define ND 64
#define ND4 16      // ND / 4

constexpr double dDX   = 2.0 / 5.0;     // 2/(B+1)
constexpr double dDY   = 2.0 / 127.0;   // 2/(H+1)
constexpr double dDX2  = dDX * dDX;
constexpr double dDY2  = dDY * dDY;
constexpr double dDEN  = 2.0 * (dDX2 + dDY2);
constexpr double dCOEF = dDX2 * dDY2 / dDEN;
constexpr double dDT   = 0.1;
constexpr double dNU   = 0.1;
constexpr double dRHO  = 1.0;
constexpr double dFRC  = 1.0;

constexpr float F_DX2    = (float)dDX2;
constexpr float F_DY2    = (float)dDY2;
constexpr float F_INVDEN = (float)(1.0 / dDEN);
constexpr float F_COEF   = (float)dCOEF;
constexpr float F_I2DX   = (float)(1.0 / (2.0 * dDX));
constexpr float F_I2DY   = (float)(1.0 / (2.0 * dDY));
constexpr float F_IDT    = (float)(1.0 / dDT);
constexpr float F_DTDX   = (float)(dDT / dDX);
constexpr float F_DTDY   = (float)(dDT / dDY);
constexpr float F_PDX    = (float)(dDT / (2.0 * dRHO * dDX));
constexpr float F_PDY    = (float)(dDT / (2.0 * dRHO * dDY));
constexpr float F_NUX    = (float)(dNU * dDT / dDX2);
constexpr float F_NUY    = (float)(dNU * dDT / dDY2);
constexpr float F_FRCDT  = (float)(dFRC * dDT);

__device__ __forceinline__ int idx4(int bb, int i, int j, int d4) {
  return ((bb * NH + i) * NW + j) * ND4 + d4;
}

// ---------------- WMMA GEMM: p = pad(x) @ W + bias ----------------
typedef __attribute__((ext_vector_type(2))) float v2f;
typedef __attribute__((ext_vector_type(8))) float v8f;

// One wave computes a 16(M) x 64(N) tile of p (M = 4*128*128 rows, K = 64).
// Border rows (padding of x) have A == 0 -> result = bias, matching reference.
__global__ __launch_bounds__(256) void gemm_p0_kernel(
    const float* __restrict__ x, const float* __restrict__ Wm,
    const float* __restrict__ bias, float* __restrict__ p) {
  const int lane  = threadIdx.x & 31;
  const int wave  = threadIdx.x >> 5;
  const int tile  = blockIdx.x * 8 + wave;       // 4096 tiles of 16 rows
  const int m0    = tile * 16;
  const int mrow  = lane & 15;
  const int khalf = lane >> 4;                   // 0 or 1
  const int m     = m0 + mrow;

  // decode global row -> (b, i, j) of the padded (4,128,128) grid
  const int bb  = m >> 14;
  const int rem = m & 16383;
  const int i   = rem >> 7;
  const int j   = rem & 127;
  const bool border = (i == 0) || (i == 127) || (j == 0) || (j == 127);

  // Preload this lane's A strip: for each k0 step of 4, lane holds K = k0+2*khalf, +1
  float fa[32];
#pragma unroll
  for (int t = 0; t < 32; ++t) fa[t] = 0.0f;
  if (!border) {
    const float* xrow = x + ((size_t)(bb * 126 + (i - 1)) * 126 + (j - 1)) * 64;
#pragma unroll
    for (int k0 = 0; k0 < 16; ++k0) {
      const int k = k0 * 4 + khalf * 2;
      fa[2 * k0]     = xrow[k];
      fa[2 * k0 + 1] = xrow[k + 1];
    }
  }

  const int ncol = lane & 15;
#pragma unroll
  for (int nt = 0; nt < 4; ++nt) {
    const int n0 = nt * 16;
    const float bval = bias[n0 + ncol];
    v8f c;
#pragma unroll
    for (int r = 0; r < 8; ++r) c[r] = bval;   // C init = bias (border rows stay = bias)
#pragma unroll
    for (int k0 = 0; k0 < 16; ++k0) {
      const int k = k0 * 4 + khalf * 2;
      v2f a; a[0] = fa[2 * k0]; a[1] = fa[2 * k0 + 1];
      v2f b; b[0] = Wm[k * 64 + n0 + ncol]; b[1] = Wm[(k + 1) * 64 + n0 + ncol];
      c = __builtin_amdgcn_wmma_f32_16x16x4_f32(false, a, false, b,
                                                (short)0, c, false, false);
    }
    // D layout: VGPR r -> row r + 8*khalf, col = n0 + (lane&15)
#pragma unroll
    for (int r = 0; r < 8; ++r) {
      const int mr = m0 + r + 8 * khalf;
      p[(size_t)mr * 64 + n0 + ncol] = c[r];
    }
  }
}

// ---------------- b-field: expr from u, v (rows 1..126, all cols) ----------------
__global__ __launch_bounds__(256) void bcomp_kernel(
    const float4* __restrict__ u, const float4* __restrict__ v,
    float4* __restrict__ bout) {
  const int t = blockIdx.x * blockDim.x + threadIdx.x;
  if (t >= NB * 126 * NW * ND4) return;
  const int d4 = t & 15;
  const int j  = (t >> 4) & 127;
  const int r  = t >> 11;            // 0..503
  const int i  = (r % 126) + 1;
  const int bb = r / 126;
  int jj = j - 1;
  if (jj < 0) jj = 0;
  if (jj > 125) jj = 125;

  const float4 ue = u[idx4(bb, i, jj + 2, d4)];
  const float4 uw = u[idx4(bb, i, jj,     d4)];
  const float4 vs = v[idx4(bb, i + 1, jj + 1, d4)];
  const float4 vn = v[idx4(bb, i - 1, jj + 1, d4)];

  float4 out;
  {
    float udx, vdy, s;
    udx = (ue.x - uw.x) * F_I2DX; vdy = (vs.x - vn.x) * F_I2DY; s = udx + vdy;
    out.x = s * F_IDT - s * s;
    udx = (ue.y - uw.y) * F_I2DX; vdy = (vs.y - vn.y) * F_I2DY; s = udx + vdy;
    out.y = s * F_IDT - s * s;
    udx = (ue.z - uw.z) * F_I2DX; vdy = (vs.z - vn.z) * F_I2DY; s = udx + vdy;
    out.z = s * F_IDT - s * s;
    udx = (ue.w - uw.w) * F_I2DX; vdy = (vs.w - vn.w) * F_I2DY; s = udx + vdy;
    out.w = s * F_IDT - s * s;
  }
  bout[idx4(bb, i, j, d4)] = out;  // RHO == 1
}

// ---------------- Jacobi pressure sweep (one pstep) ----------------
__global__ __launch_bounds__(256) void pstep_kernel(
    const float4* __restrict__ pn, const float4* __restrict__ bsrc,
    float4* __restrict__ pout) {
  const int t = blockIdx.x * blockDim.x + threadIdx.x;
  if (t >= NB * 126 * NW * ND4) return;
  const int d4 = t & 15;
  const int j  = (t >> 4) & 127;
  const int r  = t >> 11;
  const int i  = (r % 126) + 1;
  const int bb = r / 126;

  const int je = (j + 1) & 127;   // periodic wrap matches cols 0 / -1 updates
  const int jw = (j - 1) & 127;

  const float4 e  = pn[idx4(bb, i, je, d4)];
  const float4 w  = pn[idx4(bb, i, jw, d4)];
  const float4 s  = pn[idx4(bb, i + 1, j, d4)];
  const float4 n  = pn[idx4(bb, i - 1, j, d4)];
  const float4 bv = bsrc[idx4(bb, i, j, d4)];

  float4 res;
  res.x = ((e.x + w.x) * F_DY2 + (s.x + n.x) * F_DX2) * F_INVDEN - F_COEF * bv.x;
  res.y = ((e.y + w.y) * F_DY2 + (s.y + n.y) * F_DX2) * F_INVDEN - F_COEF * bv.y;
  res.z = ((e.z + w.z) * F_DY2 + (s.z + n.z) * F_DX2) * F_INVDEN - F_COEF * bv.z;
  res.w = ((e.w + w.w) * F_DY2 + (s.w + n.w) * F_DX2) * F_INVDEN - F_COEF * bv.w;

  pout[idx4(bb, i, j, d4)] = res;
  if (i == 1)   pout[idx4(bb, 0,   j, d4)] = res;  // p[:,0]  = p[:,1]  (new)
  if (i == 126) pout[idx4(bb, 127, j, d4)] = res;  // p[:,-1] = p[:,-2] (new)
}

// ---------------- momentum update ----------------
__device__ __forceinline__ float u_upd(float uc, float uw, float un_, float ue, float us,
                                       float vc, float pe, float pw) {
  return uc - uc * F_DTDX * (uc - uw) - vc * F_DTDY * (uc - un_)
         - F_PDX * (pe - pw)
         + F_NUX * (ue - 2.0f * uc + uw) + F_NUY * (us - 2.0f * uc + un_)
         + F_FRCDT;
}
__device__ __forceinline__ float v_upd(float vc, float vw, float vn_, float ve, float vs,
                                       float uc, float ps, float pn_) {
  return vc - uc * F_DTDX * (vc - vw) - vc * F_DTDY * (vc - vn_)
         - F_PDY * (ps - pn_)
         + F_NUX * (ve - 2.0f * vc + vw) + F_NUY * (vs - 2.0f * vc + vn_);
}

__global__ __launch_bounds__(256) void momentum_kernel(
    const float4* __restrict__ u, const float4* __restrict__ v,
    const float4* __restrict__ p,
    float4* __restrict__ uout, float4* __restrict__ vout) {
  const int t = blockIdx.x * blockDim.x + threadIdx.x;
  if (t >= NB * NH * NW * ND4) return;
  const int d4 = t & 15;
  const int j  = (t >> 4) & 127;
  const int i  = (t >> 11) & 127;
  const int bb = t >> 18;

  const int self = idx4(bb, i, j, d4);
  if (i == 0 || i == 127) {                  // rows zeroed every outer step
    float4 z; z.x = z.y = z.z = z.w = 0.0f;
    uout[self] = z; vout[self] = z;
    return;
  }
  if (j == 0 || j == 127) {                  // border cols never updated
    uout[self] = u[self]; vout[self] = v[self];
    return;
  }

  const float4 uc = u[self];
  const float4 uw = u[idx4(bb, i, j - 1, d4)];
  const float4 ue = u[idx4(bb, i, j + 1, d4)];
  const float4 un = u[idx4(bb, i - 1, j, d4)];
  const float4 us = u[idx4(bb, i + 1, j, d4)];
  const float4 vc = v[self];
  const float4 vw = v[idx4(bb, i, j - 1, d4)];
  const float4 ve = v[idx4(bb, i, j + 1, d4)];
  const float4 vn = v[idx4(bb, i - 1, j, d4)];
  const float4 vs = v[idx4(bb, i + 1, j, d4)];
  const float4 pe = p[idx4(bb, i, j + 1, d4)];
  const float4 pw = p[idx4(bb, i, j - 1, d4)];
  const float4 pn = p[idx4(bb, i - 1, j, d4)];
  const float4 ps = p[idx4(bb, i + 1, j, d4)];

  float4 uo, vo;
  uo.x = u_upd(uc.x, uw.x, un.x, ue.x, us.x, vc.x, pe.x, pw.x);
  uo.y = u_upd(uc.y, uw.y, un.y, ue.y, us.y, vc.y, pe.y, pw.y);
  uo.z = u_upd(uc.z, uw.z, un.z, ue.z, us.z, vc.z, pe.z, pw.z);
  uo.w = u_upd(uc.w, uw.w, un.w, ue.w, us.w, vc.w, pe.w, pw.w);
  vo.x = v_upd(vc.x, vw.x, vn.x, ve.x, vs.x, uc.x, ps.x, pn.x);
  vo.y = v_upd(vc.y, vw.y, vn.y, ve.y, vs.y, uc.y, ps.y, pn.y);
  vo.z = v_upd(vc.z, vw.z, vn.z, ve.z, vs.z, uc.z, ps.z, pn.z);
  vo.w = v_upd(vc.w, vw.w, vn.w, ve.w, vs.w, uc.w, ps.w, pn.w);
  uout[self] = uo;
  vout[self] = vo;
}

// ---------------- output: p[:, -1, :, :] ----------------
__global__ __launch_bounds__(256) void extract_kernel(
    const float4* __restrict__ p, float4* __restrict__ out) {
  const int t = blockIdx.x * blockDim.x + threadIdx.x;
  if (t >= NB * NW * ND4) return;
  const int d4 = t & 15;
  const int j  = (t >> 4) & 127;
  const int bb = t >> 11;
  out[t] = p[idx4(bb, 127, j, d4)];
}

// ---------------- host ----------------
extern "C" void kernel_launch(void* const* d_in, const int* in_sizes, int n_in,
                              void* d_out, int out_size, void* d_ws, size_t ws_size,
                              hipStream_t stream) {
  (void)in_sizes; (void)n_in; (void)out_size; (void)ws_size;
  const float* x    = (const float*)d_in[0];
  const float* Wm   = (const float*)d_in[1];
  const float* bl   = (const float*)d_in[2];
  const float* u_in = (const float*)d_in[3];
  const float* v_in = (const float*)d_in[4];

  const size_t NBUF = (size_t)NB * NH * NW * ND;     // floats per field
  float* base = (float*)d_ws;
  float* u_cur = base + 0 * NBUF;
  float* v_cur = base + 1 * NBUF;
  float* p_cur = base + 2 * NBUF;
  float* p_alt = base + 3 * NBUF;
  float* b_buf = base + 4 * NBUF;

  hipMemcpyAsync(u_cur, u_in, NBUF * sizeof(float), hipMemcpyDeviceToDevice, stream);
  hipMemcpyAsync(v_cur, v_in, NBUF * sizeof(float), hipMemcpyDeviceToDevice, stream);

  gemm_p0_kernel<<<512, 256, 0, stream>>>(x, Wm, bl, p_cur);

  const int stencilBlocks = (NB * 126 * NW * ND4) / 256;   // 4032
  const int fullBlocks    = (NB * NH * NW * ND4) / 256;    // 4096

  for (int it = 0; it < 5; ++it) {
    bcomp_kernel<<<stencilBlocks, 256, 0, stream>>>(
        (const float4*)u_cur, (const float4*)v_cur, (float4*)b_buf);
    for (int k = 0; k < 50; ++k) {
      pstep_kernel<<<stencilBlocks, 256, 0, stream>>>(
          (const float4*)p_cur, (const float4*)b_buf, (float4*)p_alt);
      float* tp = p_cur; p_cur = p_alt; p_alt = tp;
    }
    momentum_kernel<<<fullBlocks, 256, 0, stream>>>(
        (const float4*)u_cur, (const float4*)v_cur, (const float4*)p_cur,
        (float4*)p_alt, (float4*)b_buf);
    float* nu = p_alt; float* nv = b_buf;
    p_alt = u_cur; b_buf = v_cur;
    u_cur = nu; v_cur = nv;
  }

  extract_kernel<<<32, 256, 0, stream>>>((const float4*)p_cur, (float4*)d_out);
}